// VectorSelfAttention_34308198761111
// MI455X (gfx1250) — compile-verified
//
#include <hip/hip_runtime.h>
#include <hip/hip_bf16.h>
#include <math.h>

typedef float v2f __attribute__((ext_vector_type(2)));
typedef float v8f __attribute__((ext_vector_type(8)));

#define VSA_WAVES 4

// ---------------------------------------------------------------------------
// Kernel 1: mean-center q,k,v per batch; emit centered q,v and packed
// k4 = (kx, ky, kz, |k|^2).  One block per batch (B=4), 256 threads.
// ---------------------------------------------------------------------------
__global__ __launch_bounds__(256)
void vsa_center_kernel(const float* __restrict__ q, const float* __restrict__ k,
                       const float* __restrict__ v,
                       float* __restrict__ qc, float4* __restrict__ k4,
                       float* __restrict__ vc, int N) {
    const int b   = blockIdx.x;
    const int tid = threadIdx.x;
    const float* qb = q + (size_t)b * N * 3;
    const float* kb = k + (size_t)b * N * 3;
    const float* vb = v + (size_t)b * N * 3;
    float*  qcb = qc + (size_t)b * N * 3;
    float*  vcb = vc + (size_t)b * N * 3;
    float4* k4b = k4 + (size_t)b * N;

    float s[9];
#pragma unroll
    for (int c = 0; c < 9; ++c) s[c] = 0.f;

    for (int i = tid; i < N; i += blockDim.x) {
        s[0] += qb[3*i+0]; s[1] += qb[3*i+1]; s[2] += qb[3*i+2];
        s[3] += kb[3*i+0]; s[4] += kb[3*i+1]; s[5] += kb[3*i+2];
        s[6] += vb[3*i+0]; s[7] += vb[3*i+1]; s[8] += vb[3*i+2];
    }

    __shared__ float red[256];
    __shared__ float mean[9];
    for (int c = 0; c < 9; ++c) {
        red[tid] = s[c];
        __syncthreads();
        for (int off = 128; off > 0; off >>= 1) {
            if (tid < off) red[tid] += red[tid + off];
            __syncthreads();
        }
        if (tid == 0) mean[c] = red[0] / (float)N;
        __syncthreads();
    }

    for (int i = tid; i < N; i += blockDim.x) {
        float qx = qb[3*i+0] - mean[0];
        float qy = qb[3*i+1] - mean[1];
        float qz = qb[3*i+2] - mean[2];
        float kx = kb[3*i+0] - mean[3];
        float ky = kb[3*i+1] - mean[4];
        float kz = kb[3*i+2] - mean[5];
        qcb[3*i+0] = qx; qcb[3*i+1] = qy; qcb[3*i+2] = qz;
        vcb[3*i+0] = vb[3*i+0] - mean[6];
        vcb[3*i+1] = vb[3*i+1] - mean[7];
        vcb[3*i+2] = vb[3*i+2] - mean[8];
        k4b[i] = make_float4(kx, ky, kz, kx*kx + ky*ky + kz*kz);
    }
}

// ---------------------------------------------------------------------------
// Kernel 2: attention. Grid (N/16, B), 4 waves per block.
// Each wave owns the same 16-row i-tile; waves split the j-tiles 4 ways.
// Q.K^T dots via V_WMMA_F32_16X16X4_F32 (K dim = x,y,z,0).
//
// A-matrix (16x4 f32, ISA layout): lanes 0-15: M=lane, VGPR0=K0(x) VGPR1=K1(y)
//                                  lanes 16-31: M=lane-16, VGPR0=K2(z) VGPR1=K3(0)
// B-matrix (4x16 f32, rows striped across lanes): lanes 0-15: K0(x)/K1(y),
//                                  lanes 16-31: K2(z)/K3(0), N = lane&15.
// D layout: VGPR r, lanes 0-15 -> (M=r, N=lane); lanes 16-31 -> (M=r+8, N=lane-16).
// ---------------------------------------------------------------------------
__global__ __launch_bounds__(32 * VSA_WAVES)
void vsa_attn_kernel(const float* __restrict__ qc, const float4* __restrict__ k4,
                     const float* __restrict__ vc, float* __restrict__ out, int N) {
    const int b    = blockIdx.y;
    const int it   = blockIdx.x * 16;
    const int tid  = threadIdx.x;
    const int wave = tid >> 5;
    const int lane = tid & 31;
    const int half = lane >> 4;   // 0: rows 0-7 / K=0,1 ; 1: rows 8-15 / K=2,3
    const int jcol = lane & 15;

    __shared__ float sw[16], sx[16], sy[16], sz[16];
    if (tid < 16) { sw[tid] = 0.f; sx[tid] = 0.f; sy[tid] = 0.f; sz[tid] = 0.f; }
    __syncthreads();

    const float*  qb  = qc + (size_t)b * N * 3;
    const float4* k4b = k4 + (size_t)b * N;

    // Build A tile registers.
    const int arow = it + jcol;
    v2f a;
    a.x = half ? qb[3*arow+2] : qb[3*arow+0];
    a.y = half ? 0.f          : qb[3*arow+1];

    // |q|^2 for this lane's 8 rows.
    float qq8[8];
#pragma unroll
    for (int r = 0; r < 8; ++r) {
        int row = it + r + 8 * half;
        float x = qb[3*row+0], y = qb[3*row+1], z = qb[3*row+2];
        qq8[r] = x*x + y*y + z*z;
    }

    float accw[8], ax[8], ay[8], az[8];
#pragma unroll
    for (int r = 0; r < 8; ++r) { accw[r] = 0.f; ax[r] = 0.f; ay[r] = 0.f; az[r] = 0.f; }

    // exp(eta/sqrt(N)) == exp2(eta * log2(e)/sqrt(N)); fold into one constant.
    const float escale = 1.44269504088896340736f / __builtin_amdgcn_sqrtf((float)N);
    const int   ntiles = N >> 4;

    for (int tj = wave; tj < ntiles; tj += VSA_WAVES) {
        const int    j  = (tj << 4) + jcol;
        const float4 kj = k4b[j];                 // one global_load_b128

        v2f bm;
        bm.x = half ? kj.z : kj.x;
        bm.y = half ? 0.f  : kj.y;

        v8f d = {0.f, 0.f, 0.f, 0.f, 0.f, 0.f, 0.f, 0.f};
        d = __builtin_amdgcn_wmma_f32_16x16x4_f32(
                /*neg_a=*/false, a, /*neg_b=*/false, bm,
                /*c_mod=*/(short)0, d, /*reuse_a=*/false, /*reuse_b=*/false);

#pragma unroll
        for (int r = 0; r < 8; ++r) {
            float dot  = d[r];
            float eta2 = fmaxf(fmaf(qq8[r], kj.w, -dot * dot), 0.f);
            float eta  = __builtin_amdgcn_sqrtf(eta2);      // raw v_sqrt_f32
            float w    = __builtin_amdgcn_exp2f(eta * escale); // raw v_exp_f32
            accw[r] += w;
            ax[r] = fmaf(w, kj.x, ax[r]);
            ay[r] = fmaf(w, kj.y, ay[r]);
            az[r] = fmaf(w, kj.z, az[r]);
        }
    }

    // Butterfly-reduce over the 16 j-lanes of each half (masks < 16 stay in half).
#pragma unroll
    for (int m = 8; m >= 1; m >>= 1) {
#pragma unroll
        for (int r = 0; r < 8; ++r) {
            accw[r] += __shfl_xor(accw[r], m, 32);
            ax[r]   += __shfl_xor(ax[r],   m, 32);
            ay[r]   += __shfl_xor(ay[r],   m, 32);
            az[r]   += __shfl_xor(az[r],   m, 32);
        }
    }

    // Combine the 4 waves' partial sums in LDS (ds_add_f32).
    if (jcol == 0) {
#pragma unroll
        for (int r = 0; r < 8; ++r) {
            int rr = r + 8 * half;
            atomicAdd(&sw[rr], accw[r]);
            atomicAdd(&sx[rr], ax[r]);
            atomicAdd(&sy[rr], ay[r]);
            atomicAdd(&sz[rr], az[r]);
        }
    }
    __syncthreads();

    // Finalize: u = (1/N) * (q x s) x v, s = (sum w*k) / (sum w).
    if (tid < 16) {
        const int row = it + tid;
        const float* vb = vc + (size_t)b * N * 3;
        float qx = qb[3*row+0], qy = qb[3*row+1], qz = qb[3*row+2];
        float vx = vb[3*row+0], vy = vb[3*row+1], vz = vb[3*row+2];
        float inv = 1.0f / sw[tid];
        float sxv = sx[tid] * inv, syv = sy[tid] * inv, szv = sz[tid] * inv;
        // t = q x s
        float tx = qy * szv - qz * syv;
        float ty = qz * sxv - qx * szv;
        float tz = qx * syv - qy * sxv;
        // u = (t x v) / N
        float rN = 1.0f / (float)N;
        float* ob = out + (size_t)b * N * 3;
        ob[3*row+0] = (ty * vz - tz * vy) * rN;
        ob[3*row+1] = (tz * vx - tx * vz) * rN;
        ob[3*row+2] = (tx * vy - ty * vx) * rN;
    }
}

// ---------------------------------------------------------------------------
extern "C" void kernel_launch(void* const* d_in, const int* in_sizes, int n_in,
                              void* d_out, int out_size, void* d_ws, size_t ws_size,
                              hipStream_t stream) {
    const float* q = (const float*)d_in[0];
    const float* k = (const float*)d_in[1];
    const float* v = (const float*)d_in[2];

    const int B = 4;                       // per reference setup_inputs()
    const int N = in_sizes[0] / (B * 3);   // 2048

    float* ws = (float*)d_ws;
    const size_t per3 = (size_t)B * N * 3;
    float*  qc = ws;                       // B*N*3
    float*  vc = ws + per3;                // B*N*3
    float4* k4 = (float4*)(ws + 2 * per3); // B*N*4 (16B aligned: 2*per3 floats = 196608B)

    vsa_center_kernel<<<B, 256, 0, stream>>>(q, k, v, qc, k4, vc, N);

    dim3 grid(N / 16, B);
    vsa_attn_kernel<<<grid, 32 * VSA_WAVES, 0, stream>>>(qc, k4, vc,
                                                         (float*)d_out, N);
}